// SLAMv2_84928683311113
// MI455X (gfx1250) — compile-verified
//
#include <hip/hip_runtime.h>
#include <math.h>

// Problem constants (from reference)
#define NB   2      // batch
#define NS   1024   // sequence
#define ND   1000   // model dim (= NE*NHD)
#define NE   20     // experts
#define NHD  50     // head dim per expert
#define NF   2048   // ffn dim
#define NL   14     // layers
#define NCYC 3
#define CHUNK 4     // (b,e) attention pairs per score-buffer chunk

typedef __attribute__((ext_vector_type(2))) float v2f;
typedef __attribute__((ext_vector_type(8))) float v8f;

__host__ __device__ static inline int cdiv(long a, long b) { return (int)((a + b - 1) / b); }
__host__ __device__ static inline int imin(int a, int b) { return a < b ? a : b; }
__host__ __device__ static inline int imax(int a, int b) { return a > b ? a : b; }

// ---------------------------------------------------------------------------
// Generic batched WMMA f32 GEMM (templated on B orientation):
//   C[m,n] = act( alpha * sum_k A[m,k] * B'[k,n] + bias[n] )
//   OPB==0: B'[k,n] = B[k*ldb + n]   (row-major B; strided-K loads via
//           incrementing pointers -> add-chains, no v_mul_u64 in the loop)
//   OPB==1: B'[k,n] = B[n*ldb + k]   (transposed B, Q@K^T; contiguous-K loads
//           -> global_load_b64 with immediate offsets, same as A)
//
// One wave computes a 32x32 C tile (4 v_wmma_f32_16x16x4_f32 accumulators).
// K is unrolled by 16 (16 WMMAs per main-loop iteration). Edge tiles are
// handled by CLAMPING load rows/cols into range: in a 16x16 WMMA, A row m
// feeds only C row m and B col n only C col n, so clamped (garbage) lanes
// only affect C entries that the guarded store never writes. Only the K
// dimension needs true zero-padding (short tail loop).
// Tile index is wave-uniform -> EXEC is all-ones at each WMMA (ISA req).
// ---------------------------------------------------------------------------
template <int OPB>
__global__ __launch_bounds__(128)
void wmma_gemm_f32(const float* __restrict__ A, const float* __restrict__ Bm,
                   const float* __restrict__ bias, float* __restrict__ C,
                   int M, int N, int K, int lda, int ldb, int ldc,
                   long sA, long sB, long sC, long sBias,
                   float alpha, int act)
{
  const int batch = blockIdx.y;
  A  += (long)batch * sA;
  Bm += (long)batch * sB;
  C  += (long)batch * sC;
  const float* bp = bias ? (bias + (long)batch * sBias) : nullptr;

  const int tilesN = (N + 31) >> 5;
  const int tilesM = (M + 31) >> 5;
  const int wave = threadIdx.x >> 5;
  const int lane = threadIdx.x & 31;
  const int tile = blockIdx.x * 4 + wave;
  if (tile >= tilesM * tilesN) return;
  const int tm = tile / tilesN;
  const int tn = tile - tm * tilesN;
  const int row0 = tm << 5, col0 = tn << 5;

  // Fragment layouts (CDNA5 ISA, 32-bit 16x4 / 4x16 / 16x16):
  //  A: lanes 0-15 -> M=lane, (x,y)=(K0,K1); lanes 16-31 -> (K2,K3)
  //  B: lanes 0-15 -> N=lane, (x,y)=(K0,K1); lanes 16-31 -> (K2,K3)
  //  C/D: VGPR r: lanes 0-15 -> M=r, N=lane; lanes 16-31 -> M=r+8, N=lane-16
  const int half = lane >> 4;    // 0 or 1 -> K sub-pair
  const int l    = lane & 15;

  // Clamped load rows/cols (always in-bounds; see header comment)
  const int ar0 = imin(row0 + l,      M - 1);
  const int ar1 = imin(row0 + 16 + l, M - 1);
  const int bc0 = imin(col0 + l,      N - 1);
  const int bc1 = imin(col0 + 16 + l, N - 1);

  const float* Ap0 = A + (long)ar0 * lda;
  const float* Ap1 = A + (long)ar1 * lda;

  const long ldbL = ldb;
  // OPB==1: row-base pointers (K contiguous). OPB==0: column-base pointers.
  const float* Bp0 = Bm + (long)bc0 * (OPB ? ldbL : 1L);
  const float* Bp1 = Bm + (long)bc1 * (OPB ? ldbL : 1L);

  v8f acc00 = {0.f,0.f,0.f,0.f,0.f,0.f,0.f,0.f};
  v8f acc01 = acc00, acc10 = acc00, acc11 = acc00;

  const int koff = 2 * half;             // this lane's K sub-offset in a frag
  const int Kmain = K & ~15;

  // Incrementing B base pointers for OPB==0 (advance 16 rows of B per iter)
  const float* Bk0 = Bp0 + (long)koff * (OPB ? 1L : ldbL);
  const float* Bk1 = Bp1 + (long)koff * (OPB ? 1L : ldbL);

  for (int k0 = 0; k0 < Kmain; k0 += 16) {
    // Prefetch next K block of A rows (gfx1250 global_prefetch_b8)
    __builtin_prefetch(Ap0 + k0 + 16, 0, 1);
    __builtin_prefetch(Ap1 + k0 + 16, 0, 1);

    // Issue ALL fragment loads for this iteration first, then 16 WMMAs,
    // so the compiler can batch waits and pipeline loads under WMMAs.
    v2f a0[4], a1[4], b0[4], b1[4];
#pragma unroll
    for (int i = 0; i < 4; ++i) {
      const int ka = k0 + 4 * i + koff;
      a0[i].x = Ap0[ka];     a0[i].y = Ap0[ka + 1];
      a1[i].x = Ap1[ka];     a1[i].y = Ap1[ka + 1];
      if (OPB) {  // contiguous along K -> merged b64 with immediate offsets
        b0[i].x = Bp0[ka];   b0[i].y = Bp0[ka + 1];
        b1[i].x = Bp1[ka];   b1[i].y = Bp1[ka + 1];
      } else {    // strided along K -> pointer add-chains only
        const float* q0 = Bk0 + (long)(4 * i) * ldbL;
        const float* q1 = Bk1 + (long)(4 * i) * ldbL;
        b0[i].x = q0[0];     b0[i].y = q0[ldbL];
        b1[i].x = q1[0];     b1[i].y = q1[ldbL];
      }
    }
#pragma unroll
    for (int i = 0; i < 4; ++i) {
      acc00 = __builtin_amdgcn_wmma_f32_16x16x4_f32(false, a0[i], false, b0[i],
                                                    (short)0, acc00, false, false);
      acc01 = __builtin_amdgcn_wmma_f32_16x16x4_f32(false, a0[i], false, b1[i],
                                                    (short)0, acc01, false, false);
      acc10 = __builtin_amdgcn_wmma_f32_16x16x4_f32(false, a1[i], false, b0[i],
                                                    (short)0, acc10, false, false);
      acc11 = __builtin_amdgcn_wmma_f32_16x16x4_f32(false, a1[i], false, b1[i],
                                                    (short)0, acc11, false, false);
    }
    if (!OPB) { Bk0 += 16L * ldbL; Bk1 += 16L * ldbL; }
  }

  // K tail (K % 16 != 0): zero-pad along K with guarded loads.
  for (int k0 = Kmain; k0 < K; k0 += 4) {
    const int ka = k0 + koff;
    const long bK = OPB ? 1L : ldbL;
    v2f a0v, a1v, b0v, b1v;
    a0v.x = (ka     < K) ? Ap0[ka]     : 0.f;
    a0v.y = (ka + 1 < K) ? Ap0[ka + 1] : 0.f;
    a1v.x = (ka     < K) ? Ap1[ka]     : 0.f;
    a1v.y = (ka + 1 < K) ? Ap1[ka + 1] : 0.f;
    b0v.x = (ka     < K) ? Bp0[(long)ka * bK]       : 0.f;
    b0v.y = (ka + 1 < K) ? Bp0[(long)(ka + 1) * bK] : 0.f;
    b1v.x = (ka     < K) ? Bp1[(long)ka * bK]       : 0.f;
    b1v.y = (ka + 1 < K) ? Bp1[(long)(ka + 1) * bK] : 0.f;
    acc00 = __builtin_amdgcn_wmma_f32_16x16x4_f32(false, a0v, false, b0v,
                                                  (short)0, acc00, false, false);
    acc01 = __builtin_amdgcn_wmma_f32_16x16x4_f32(false, a0v, false, b1v,
                                                  (short)0, acc01, false, false);
    acc10 = __builtin_amdgcn_wmma_f32_16x16x4_f32(false, a1v, false, b0v,
                                                  (short)0, acc10, false, false);
    acc11 = __builtin_amdgcn_wmma_f32_16x16x4_f32(false, a1v, false, b1v,
                                                  (short)0, acc11, false, false);
  }

  // Store 4 sub-tiles with bounds guards.
  const int ccol0 = col0 + l;
  const int ccol1 = col0 + 16 + l;
  const float bv0 = (bp && ccol0 < N) ? bp[ccol0] : 0.f;
  const float bv1 = (bp && ccol1 < N) ? bp[ccol1] : 0.f;
#pragma unroll
  for (int r = 0; r < 8; ++r) {
    const int crow0 = row0 + half * 8 + r;
    const int crow1 = crow0 + 16;
    if (crow0 < M && ccol0 < N) {
      float v = acc00[r] * alpha + bv0;
      if (act == 1) v = 0.5f * v * (1.0f + erff(v * 0.70710678118654752f));
      C[(long)crow0 * ldc + ccol0] = v;
    }
    if (crow0 < M && ccol1 < N) {
      float v = acc01[r] * alpha + bv1;
      if (act == 1) v = 0.5f * v * (1.0f + erff(v * 0.70710678118654752f));
      C[(long)crow0 * ldc + ccol1] = v;
    }
    if (crow1 < M && ccol0 < N) {
      float v = acc10[r] * alpha + bv0;
      if (act == 1) v = 0.5f * v * (1.0f + erff(v * 0.70710678118654752f));
      C[(long)crow1 * ldc + ccol0] = v;
    }
    if (crow1 < M && ccol1 < N) {
      float v = acc11[r] * alpha + bv1;
      if (act == 1) v = 0.5f * v * (1.0f + erff(v * 0.70710678118654752f));
      C[(long)crow1 * ldc + ccol1] = v;
    }
  }
}

// ---------------------------------------------------------------------------
// Row softmax (in place). grid.x = rows, 256 threads per row.
// ---------------------------------------------------------------------------
__global__ __launch_bounds__(256)
void softmax_rows(float* __restrict__ X, int cols, int ld)
{
  float* x = X + (long)blockIdx.x * ld;
  __shared__ float red[256];
  const int tid = threadIdx.x;

  float m = -3.4e38f;
  for (int c = tid; c < cols; c += 256) m = fmaxf(m, x[c]);
  red[tid] = m; __syncthreads();
  for (int s = 128; s > 0; s >>= 1) {
    if (tid < s) red[tid] = fmaxf(red[tid], red[tid + s]);
    __syncthreads();
  }
  m = red[0]; __syncthreads();

  float sum = 0.f;
  for (int c = tid; c < cols; c += 256) {
    float e = expf(x[c] - m);
    x[c] = e;
    sum += e;
  }
  red[tid] = sum; __syncthreads();
  for (int s = 128; s > 0; s >>= 1) {
    if (tid < s) red[tid] += red[tid + s];
    __syncthreads();
  }
  const float inv = 1.0f / red[0];
  for (int c = tid; c < cols; c += 256) x[c] *= inv;
}

// ---------------------------------------------------------------------------
// y = LayerNorm(x + r) * g + b   (grid.x = rows, 256 threads)
// ---------------------------------------------------------------------------
__global__ __launch_bounds__(256)
void add_layernorm(const float* __restrict__ X, const float* __restrict__ R,
                   const float* __restrict__ G, const float* __restrict__ Bb,
                   float* __restrict__ Y, int D)
{
  const long row = blockIdx.x;
  const float* x = X + row * D;
  const float* r = R + row * D;
  float* y = Y + row * D;
  __shared__ float s1[256], s2[256];
  const int tid = threadIdx.x;

  float a = 0.f, b = 0.f;
  for (int c = tid; c < D; c += 256) {
    float v = x[c] + r[c];
    a += v; b += v * v;
  }
  s1[tid] = a; s2[tid] = b; __syncthreads();
  for (int s = 128; s > 0; s >>= 1) {
    if (tid < s) { s1[tid] += s1[tid + s]; s2[tid] += s2[tid + s]; }
    __syncthreads();
  }
  const float mean = s1[0] / (float)D;
  const float var  = fmaxf(s2[0] / (float)D - mean * mean, 0.f);
  const float rstd = rsqrtf(var + 1e-5f);
  for (int c = tid; c < D; c += 256)
    y[c] = (x[c] + r[c] - mean) * rstd * G[c] + Bb[c];
}

// ---------------------------------------------------------------------------
// attnout[b, t, e*HD + h] = expout[(e*NB+b), t, h] * routing[b, t, e]
// ---------------------------------------------------------------------------
__global__ __launch_bounds__(256)
void combine_experts(const float* __restrict__ expout,
                     const float* __restrict__ routing,
                     float* __restrict__ attnout, int s)
{
  const long idx = (long)blockIdx.x * 256 + threadIdx.x;
  const long tot = (long)NB * s * NE * NHD;
  if (idx >= tot) return;
  const int h = (int)(idx % NHD);
  long t1 = idx / NHD;
  const int e = (int)(t1 % NE);
  t1 /= NE;
  const int t = (int)(t1 % s);
  const int b = (int)(t1 / s);
  const float v = expout[((long)(e * NB + b) * s + t) * NHD + h];
  const float w = routing[((long)b * s + t) * NE + e];
  attnout[((long)b * s + t) * ND + e * NHD + h] = v * w;
}

// ---------------------------------------------------------------------------
// SLAM segment plumbing
// ---------------------------------------------------------------------------
__global__ __launch_bounds__(256)
void gather_seg(const float* __restrict__ X, float* __restrict__ Y,
                int st, int slen)
{
  const long idx = (long)blockIdx.x * 256 + threadIdx.x;
  const long tot = (long)NB * slen * ND;
  if (idx >= tot) return;
  const int d = (int)(idx % ND);
  long t1 = idx / ND;
  const int i = (int)(t1 % slen);
  const int b = (int)(t1 / slen);
  int srcT = st + i; if (srcT >= NS) srcT -= NS;  // wrap for segment 3
  Y[((long)b * slen + i) * ND + d] = X[((long)b * NS + srcT) * ND + d];
}

__global__ __launch_bounds__(256)
void scatter_add_seg(const float* __restrict__ Y, float* __restrict__ Xnew,
                     int st, int slen)
{
  const long idx = (long)blockIdx.x * 256 + threadIdx.x;
  const long tot = (long)NB * slen * ND;
  if (idx >= tot) return;
  const int d = (int)(idx % ND);
  long t1 = idx / ND;
  const int i = (int)(t1 % slen);
  const int b = (int)(t1 / slen);
  int dstT = st + i; if (dstT >= NS) dstT -= NS;
  Xnew[((long)b * NS + dstT) * ND + d] += Y[((long)b * slen + i) * ND + d];
}

__global__ __launch_bounds__(256)
void finalize_cycle(const float* __restrict__ Xnew, float* __restrict__ X,
                    int st0, int en0, int st1, int en1,
                    int st2, int en2, int st3, int en3, int wrap)
{
  const long idx = (long)blockIdx.x * 256 + threadIdx.x;
  const long tot = (long)NB * NS * ND;
  if (idx >= tot) return;
  const int t = (int)((idx / ND) % NS);
  int cnt = 0;
  cnt += (t >= st0 && t < en0);
  cnt += (t >= st1 && t < en1);
  cnt += (t >= st2 && t < en2);
  cnt += (t >= st3 && t < en3);
  cnt += (t < wrap);
  if (cnt < 1) cnt = 1;
  X[idx] = Xnew[idx] / (float)cnt;
}

// ---------------------------------------------------------------------------
// Host-side orchestration
// ---------------------------------------------------------------------------
struct SlamParams {
  const float *ew, *eb, *rw, *rb, *ow, *ob;
  const float *g1, *b1, *w1, *fb1, *w2, *fb2, *g2, *b2;
};
struct SlamWs {
  float *routing, *qkv, *scores, *expout, *attnout, *tmpD, *xln, *ffh;
};

static void gemm(hipStream_t st, const float* A, const float* B,
                 const float* bias, float* C,
                 int M, int N, int K, int lda, int ldb, int ldc,
                 long sA, long sB, long sC, long sBias, int batch,
                 float alpha, int opB, int act)
{
  const long tiles = (long)cdiv(M, 32) * cdiv(N, 32);
  dim3 grid((unsigned)cdiv(tiles, 4), (unsigned)batch, 1);
  if (opB)
    wmma_gemm_f32<1><<<grid, 128, 0, st>>>(A, B, bias, C, M, N, K, lda, ldb,
                                           ldc, sA, sB, sC, sBias, alpha, act);
  else
    wmma_gemm_f32<0><<<grid, 128, 0, st>>>(A, B, bias, C, M, N, K, lda, ldb,
                                           ldc, sA, sB, sC, sBias, alpha, act);
}

// One transformer block on a contiguous [NB, s, ND] buffer.
// `out` may alias `xin` (xin is last read before `out` is written).
static void run_block(hipStream_t st, const SlamParams& P, const SlamWs& W,
                      int layer, const float* xin, float* out, int s)
{
  const int rows = NB * s;
  const int H3 = 3 * NHD;  // 150

  // 1) routing = softmax(x @ rw + rb)
  gemm(st, xin, P.rw + (long)layer * ND * NE, P.rb + (long)layer * NE,
       W.routing, rows, NE, ND, ND, NE, NE, 0, 0, 0, 0, 1, 1.f, 0, 0);
  softmax_rows<<<rows, 256, 0, st>>>(W.routing, NE, NE);

  // 2) qkv[e, row, 150] = x @ ew[layer, e] + eb[layer, e]   (batched over E)
  gemm(st, xin, P.ew + (long)layer * NE * ND * H3,
       P.eb + (long)layer * NE * H3, W.qkv,
       rows, H3, ND, ND, H3, H3,
       0, (long)ND * H3, (long)rows * H3, H3, NE, 1.f, 0, 0);

  // 3) attention per (e,b) pair, chunked to bound score-buffer size
  const int pairs = NE * NB;
  const float iscale = 0.14142135623730951f;  // HD^-0.5
  for (int p0 = 0; p0 < pairs; p0 += CHUNK) {
    const int pc = imin(CHUNK, pairs - p0);
    const float* qb = W.qkv + (long)p0 * s * H3;      // pair-major base
    // scores = (Q @ K^T) * HD^-0.5
    gemm(st, qb /*Q*/, qb + NHD /*K (opB=1)*/, nullptr, W.scores,
         s, s, NHD, H3, H3, s,
         (long)s * H3, (long)s * H3, (long)s * s, 0, pc, iscale, 1, 0);
    softmax_rows<<<pc * s, 256, 0, st>>>(W.scores, s, s);
    // expout = attn @ V
    gemm(st, W.scores, qb + 2 * NHD /*V*/, nullptr,
         W.expout + (long)p0 * s * NHD,
         s, NHD, s, s, H3, NHD,
         (long)s * s, (long)s * H3, (long)s * NHD, 0, pc, 1.f, 0, 0);
  }

  // 4) weighted head-slice combine -> attnout [rows, ND]
  {
    const long tot = (long)rows * ND;
    combine_experts<<<cdiv(tot, 256), 256, 0, st>>>(W.expout, W.routing,
                                                    W.attnout, s);
  }

  // 5) output projection
  gemm(st, W.attnout, P.ow + (long)layer * ND * ND, P.ob + (long)layer * ND,
       W.tmpD, rows, ND, ND, ND, ND, ND, 0, 0, 0, 0, 1, 1.f, 0, 0);

  // 6) x = LN1(xin + attn_out)
  add_layernorm<<<rows, 256, 0, st>>>(xin, W.tmpD, P.g1 + (long)layer * ND,
                                      P.b1 + (long)layer * ND, W.xln, ND);

  // 7) FFN: gelu(x @ w1 + b1) @ w2 + b2
  gemm(st, W.xln, P.w1 + (long)layer * ND * NF, P.fb1 + (long)layer * NF,
       W.ffh, rows, NF, ND, ND, NF, NF, 0, 0, 0, 0, 1, 1.f, 0, /*gelu*/1);
  gemm(st, W.ffh, P.w2 + (long)layer * NF * ND, P.fb2 + (long)layer * ND,
       W.tmpD, rows, ND, NF, NF, ND, ND, 0, 0, 0, 0, 1, 1.f, 0, 0);

  // 8) out = LN2(x + ffn)
  add_layernorm<<<rows, 256, 0, st>>>(W.xln, W.tmpD, P.g2 + (long)layer * ND,
                                      P.b2 + (long)layer * ND, out, ND);
}

extern "C" void kernel_launch(void* const* d_in, const int* in_sizes, int n_in,
                              void* d_out, int out_size, void* d_ws,
                              size_t ws_size, hipStream_t stream)
{
  (void)in_sizes; (void)n_in; (void)out_size; (void)ws_size;
  const float* x_in = (const float*)d_in[0];
  SlamParams P;
  P.ew  = (const float*)d_in[1];  P.eb  = (const float*)d_in[2];
  P.rw  = (const float*)d_in[3];  P.rb  = (const float*)d_in[4];
  P.ow  = (const float*)d_in[5];  P.ob  = (const float*)d_in[6];
  P.g1  = (const float*)d_in[7];  P.b1  = (const float*)d_in[8];
  P.w1  = (const float*)d_in[9];  P.fb1 = (const float*)d_in[10];
  P.w2  = (const float*)d_in[11]; P.fb2 = (const float*)d_in[12];
  P.g2  = (const float*)d_in[13]; P.b2  = (const float*)d_in[14];

  // Carve workspace (256B-aligned slices)
  char* wp = (char*)d_ws;
  auto take = [&](size_t elems) -> float* {
    float* r = (float*)wp;
    wp += ((elems * sizeof(float) + 255) / 256) * 256;
    return r;
  };
  float* xA   = take((size_t)NB * NS * ND);
  float* xnew = take((size_t)NB * NS * ND);
  float* xseg = take((size_t)NB * NS * ND);
  SlamWs W;
  W.routing = take((size_t)NB * NS * NE);
  W.qkv     = take((size_t)NE * NB * NS * 3 * NHD);
  W.scores  = take((size_t)CHUNK * NS * NS);
  W.expout  = take((size_t)NE * NB * NS * NHD);
  W.attnout = take((size_t)NB * NS * ND);
  W.tmpD    = take((size_t)NB * NS * ND);
  W.xln     = take((size_t)NB * NS * ND);
  W.ffh     = take((size_t)NB * NS * NF);

  (void)hipMemcpyAsync(xA, x_in, sizeof(float) * NB * NS * ND,
                       hipMemcpyDeviceToDevice, stream);

  // First block on full sequence (in place: xA -> xA)
  run_block(stream, P, W, 0, xA, xA, NS);

  // SLAM segment schedule (same formulas as reference, S=1024)
  const int b45 = (int)(NS * 0.45), b25 = (int)(NS * 0.25);
  const int b65 = (int)(NS * 0.65), b85 = (int)(NS * 0.85);
  int st[4] = {0, imax(b25, 0), imax(b45, 0), imax(b65, 0)};
  int en[4] = {imin(b45, NS), imin(b65, NS), imin(b85, NS), NS};
  const int wrap = (NS > 20) ? imax(imin((int)(NS * 0.05), NS / 10), 1) : 0;

  for (int c = 0; c < NCYC; ++c) {
    (void)hipMemsetAsync(xnew, 0, sizeof(float) * NB * NS * ND, stream);
    for (int j = 0; j < 4; ++j) {
      const int li = 1 + 4 * c + j;
      const int slen = (j == 3) ? (NS - st[3]) + wrap : (en[j] - st[j]);
      const long tot = (long)NB * slen * ND;
      gather_seg<<<cdiv(tot, 256), 256, 0, stream>>>(xA, xseg, st[j], slen);
      run_block(stream, P, W, li, xseg, xseg, slen);
      scatter_add_seg<<<cdiv(tot, 256), 256, 0, stream>>>(xseg, xnew,
                                                          st[j], slen);
    }
    const long tot = (long)NB * NS * ND;
    finalize_cycle<<<cdiv(tot, 256), 256, 0, stream>>>(
        xnew, xA, st[0], en[0], st[1], en[1], st[2], en[2], st[3], en[3],
        wrap);
  }

  // Final block writes straight into d_out
  run_block(stream, P, W, NL - 1, xA, (float*)d_out, NS);
}